// HingeLoss_17351667875977
// MI455X (gfx1250) — compile-verified
//
#include <hip/hip_runtime.h>
#include <hip/hip_bf16.h>

typedef float v2f __attribute__((ext_vector_type(2)));
typedef float v8f __attribute__((ext_vector_type(8)));

#define DIM 64
#define DELTA 1.0f

// Order-preserving map float -> uint so unsigned atomicMax == float max.
__device__ __forceinline__ unsigned mapf(float f) {
    unsigned u = __float_as_uint(f);
    return (u & 0x80000000u) ? ~u : (u | 0x80000000u);
}
__device__ __forceinline__ float unmapf(unsigned u) {
    unsigned b = (u & 0x80000000u) ? (u & 0x7FFFFFFFu) : ~u;
    return __uint_as_float(b);
}

// One wave handles 16 edges. fp32 WMMA 16x16x4, K swept 0..60:
//   A[m][k] = embA[inv[src[e_m]]][k]   (pc rows)
//   B[k][n] = embB[inv[dst[e_n]]][k]   (pp rows, transposed)
// diag(A*B) = per-edge dot products, full fp32 precision.
// Per ISA layout (16x4 f32 A / 4x16 f32 B), lane L supplies a float2 from
// row ids[L%16] at column k0 + (L>=16 ? 2 : 0) for BOTH operands.
// MODE 0: scatter-add (sum + count).  MODE 1: scatter-max (mapped bits) + count.
template <int MODE>
__global__ __launch_bounds__(256) void edge_dot_scatter(
    const float* __restrict__ embA, const float* __restrict__ embB,
    const int* __restrict__ inv,
    const int* __restrict__ srcIds, const int* __restrict__ dstIds,
    int nEdges,
    float* __restrict__ segSum, float* __restrict__ segCnt,
    unsigned* __restrict__ segMax)
{
    const int lane = threadIdx.x & 31;
    const long wave = (long)blockIdx.x * (blockDim.x >> 5) + (threadIdx.x >> 5);
    const long base = wave * 16;
    const int m = lane & 15;               // edge slot within wave
    long e = base + m;
    const bool valid = (e < (long)nEdges);
    const long ec = valid ? e : (long)(nEdges - 1);   // clamp: keep loads legal, EXEC full

    const int s = srcIds[ec];              // segment key (original node id)
    const int d = dstIds[ec];
    const float* rowA = embA + (long)inv[s] * DIM;
    const float* rowB = embB + (long)inv[d] * DIM;
    const int koff = (lane >> 4) << 1;     // 0 for lanes 0-15, 2 for lanes 16-31

    v8f acc = {0.f, 0.f, 0.f, 0.f, 0.f, 0.f, 0.f, 0.f};
#pragma unroll
    for (int k0 = 0; k0 < DIM; k0 += 4) {
        v2f a = *(const v2f*)(rowA + k0 + koff);
        v2f b = *(const v2f*)(rowB + k0 + koff);
        // 8 args: (neg_a, A, neg_b, B, c_mod, C, reuse_a, reuse_b)
        acc = __builtin_amdgcn_wmma_f32_16x16x4_f32(
            false, a, false, b, (short)0, acc, false, false);
    }

    // Diagonal element m lives in lane (m<8 ? m : m+16), component m&7.
    // Those "holder" lanes already carry edge m's segment key s and validity.
    const bool holder = (lane < 8) || (lane >= 24);
    const int c = lane & 7;
    float v01 = (c & 1) ? acc[1] : acc[0];
    float v23 = (c & 1) ? acc[3] : acc[2];
    float v45 = (c & 1) ? acc[5] : acc[4];
    float v67 = (c & 1) ? acc[7] : acc[6];
    float v0123 = (c & 2) ? v23 : v01;
    float v4567 = (c & 2) ? v67 : v45;
    float dot = (c & 4) ? v4567 : v0123;

    if (holder && valid) {
        if (MODE == 0) {
            atomicAdd(&segSum[s], dot);
            atomicAdd(&segCnt[s], 1.0f);
        } else {
            atomicMax(&segMax[s], mapf(dot));
            atomicAdd(&segCnt[s], 1.0f);
        }
    }
}

__global__ __launch_bounds__(256) void finalize_loss(
    const float* __restrict__ pSum, const float* __restrict__ pCnt,
    const unsigned* __restrict__ nMax, const float* __restrict__ nCnt,
    int nNodes, float* __restrict__ out)
{
    const int i = blockIdx.x * blockDim.x + threadIdx.x;
    float l = 0.0f;
    if (i < nNodes) {
        float pc = pCnt[i];
        float pd = (pc > 0.0f) ? (pSum[i] / pc) : 0.0f;
        float nc = nCnt[i];
        float nd = (nc > 0.0f) ? unmapf(nMax[i]) : 0.0f;
        l = fmaxf(nd - pd + DELTA, 0.0f);
    }
    __shared__ float red[256];
    red[threadIdx.x] = l;
    __syncthreads();
#pragma unroll
    for (int off = 128; off > 0; off >>= 1) {
        if ((int)threadIdx.x < off) red[threadIdx.x] += red[threadIdx.x + off];
        __syncthreads();
    }
    if (threadIdx.x == 0) atomicAdd(out, red[0] * (1.0f / (float)nNodes));
}

extern "C" void kernel_launch(void* const* d_in, const int* in_sizes, int n_in,
                              void* d_out, int out_size, void* d_ws, size_t ws_size,
                              hipStream_t stream) {
    const float* embSrc = (const float*)d_in[0];
    const float* embDst = (const float*)d_in[1];
    const int*   inv    = (const int*)d_in[2];
    const int*   posSrc = (const int*)d_in[3];
    const int*   posDst = (const int*)d_in[4];
    const int*   negSrc = (const int*)d_in[5];
    const int*   negDst = (const int*)d_in[6];
    const int nNodes = in_sizes[2];
    const int nEdges = in_sizes[3];

    float*    pSum = (float*)d_ws;
    float*    pCnt = pSum + nNodes;
    unsigned* nMax = (unsigned*)(pCnt + nNodes);
    float*    nCnt = (float*)(nMax + nNodes);

    hipMemsetAsync(d_ws, 0, (size_t)nNodes * 4 * sizeof(float), stream);
    hipMemsetAsync(d_out, 0, sizeof(float), stream);

    const long waves  = ((long)nEdges + 15) / 16;
    const int  wavesPerBlock = 8;   // 256 threads = 8 wave32
    const long blocks = (waves + wavesPerBlock - 1) / wavesPerBlock;

    edge_dot_scatter<0><<<blocks, 256, 0, stream>>>(
        embSrc, embDst, inv, posSrc, posDst, nEdges, pSum, pCnt, nullptr);
    edge_dot_scatter<1><<<blocks, 256, 0, stream>>>(
        embSrc, embDst, inv, negSrc, negDst, nEdges, nullptr, nCnt, nMax);

    finalize_loss<<<(nNodes + 255) / 256, 256, 0, stream>>>(
        pSum, pCnt, nMax, nCnt, nNodes, (float*)d_out);
}